// UserItemWithRepostEmbedding_70755291234320
// MI455X (gfx1250) — compile-verified
//
#include <hip/hip_runtime.h>
#include <hip/hip_bf16.h>

#define NTOKEN  100000
#define NINP    64
#define NHID    128
#define N_EDGES 600000
#define BATCH   64
#define MAX_LEN 200

typedef float v2f __attribute__((ext_vector_type(2)));
typedef float v8f __attribute__((ext_vector_type(8)));

// ---------------- degree / normalization ----------------

__global__ void deg_init_kernel(float* __restrict__ deg, int n) {
    int i = blockIdx.x * blockDim.x + threadIdx.x;
    if (i < n) deg[i] = 1.0f;                 // self-loop contribution
}

__global__ void deg_accum_kernel(const int* __restrict__ dst, float* __restrict__ deg, int ne) {
    int i = blockIdx.x * blockDim.x + threadIdx.x;
    if (i < ne) atomicAdd(&deg[dst[i]], 1.0f);
}

__global__ void dinv_kernel(float* __restrict__ deg, int n) {
    int i = blockIdx.x * blockDim.x + threadIdx.x;
    if (i < n) deg[i] = rsqrtf(deg[i]);       // deg >= 1 always (self loops added)
}

// ---------------- dense GEMM on the fp32 WMMA pipe ----------------
// Y[M x NOUT] = X[M x K] @ W[K x NOUT], row-major.
// One wave computes one 16x16 tile using V_WMMA_F32_16X16X4_F32.
// Fragment layouts per CDNA5 ISA 7.12.2:
//   A (16x4, MxK): element (m,k) -> lane m + 16*(k>>1), vgpr k&1
//   B (4x16, KxN): element (k,n) -> lane n + 16*(k>>1), vgpr k&1
//   C/D (16x16):   element (m,n) -> vgpr m%8, lane n + 16*(m/8)
template <int K, int NOUT>
__global__ __launch_bounds__(256) void gemm_wmma_f32(const float* __restrict__ X,
                                                     const float* __restrict__ W,
                                                     float* __restrict__ Y) {
    const int lane = threadIdx.x;             // 0..31
    const int half = lane >> 4;               // 0 or 1
    const int l16  = lane & 15;
    const int m0   = blockIdx.x * 16;         // row tile
    const int n0   = threadIdx.y * 16;        // col tile (wave id within block)

    const long rowA = (long)(m0 + l16);
    v8f c = {};
#pragma unroll
    for (int k0 = 0; k0 < K; k0 += 4) {
        const int ka = k0 + 2 * half;
        v2f a, b;
        a.x = X[rowA * K + ka];
        a.y = X[rowA * K + ka + 1];
        b.x = W[(long)ka * NOUT + n0 + l16];
        b.y = W[(long)(ka + 1) * NOUT + n0 + l16];
        c = __builtin_amdgcn_wmma_f32_16x16x4_f32(
                /*neg_a=*/false, a, /*neg_b=*/false, b,
                /*c_mod=*/(short)0, c, /*reuse_a=*/false, /*reuse_b=*/false);
    }
    const int ncol = n0 + l16;
#pragma unroll
    for (int r = 0; r < 8; ++r) {
        const int m = m0 + r + 8 * half;
        Y[(long)m * NOUT + ncol] = c[r];
    }
}

// ---------------- self-loop init: out = dinv^2 * h + bias ----------------

__global__ void init_out_kernel(const float* __restrict__ H, const float* __restrict__ dinv,
                                const float* __restrict__ bias, float* __restrict__ Out,
                                int n, int F) {
    long i = (long)blockIdx.x * blockDim.x + threadIdx.x;
    long total = (long)n * F;
    if (i >= total) return;
    int node = (int)(i / F);
    int j    = (int)(i % F);
    float di = dinv[node];
    Out[i] = di * di * H[i] + bias[j];
}

// ---------------- edge scatter: one wave per edge ----------------

__global__ void scatter_add_kernel(const float* __restrict__ H, const float* __restrict__ dinv,
                                   const int* __restrict__ src, const int* __restrict__ dst,
                                   float* __restrict__ Out, int ne, int F) {
    int wave = (int)(((long)blockIdx.x * blockDim.x + threadIdx.x) >> 5);
    int lane = threadIdx.x & 31;
    if (wave >= ne) return;
    int s = src[wave];
    int d = dst[wave];
    float norm = dinv[s] * dinv[d];
    const float* hp = H + (long)s * F;
    float*       op = Out + (long)d * F;
    for (int j = lane; j < F; j += 32)
        atomicAdd(&op[j], norm * hp[j]);
}

// ---------------- final row gather ----------------

__global__ void gather_rows_kernel(const float* __restrict__ Table, const int* __restrict__ idx,
                                   float* __restrict__ Out, int nrows, int F) {
    int row = blockIdx.x;
    int j   = threadIdx.x;
    if (row < nrows && j < F)
        Out[(long)row * F + j] = Table[(long)idx[row] * F + j];
}

// ---------------- host-side orchestration ----------------

extern "C" void kernel_launch(void* const* d_in, const int* in_sizes, int n_in,
                              void* d_out, int out_size, void* d_ws, size_t ws_size,
                              hipStream_t stream) {
    // setup_inputs order: input, emb, W1, b1, W2, b2, edge_index
    const int*   inp  = (const int*)  d_in[0];   // [64, 200]
    const float* emb  = (const float*)d_in[1];   // [100000, 64]
    const float* W1   = (const float*)d_in[2];   // [64, 128]
    const float* b1   = (const float*)d_in[3];   // [128]
    const float* W2   = (const float*)d_in[4];   // [128, 64]
    const float* b2   = (const float*)d_in[5];   // [64]
    const int*   ei   = (const int*)  d_in[6];   // [2, 600000]
    const int*   esrc = ei;
    const int*   edst = ei + N_EDGES;
    float*       out  = (float*)d_out;           // [64, 200, 64]

    // Workspace layout (256B-aligned):
    //   dinv : NTOKEN floats                 (0.4 MB)
    //   H    : NTOKEN*NHID floats            (51.2 MB)  (h0, later reused as h1)
    //   OUT  : NTOKEN*NHID floats            (51.2 MB)  (out1, later reused as out2)
    char* ws = (char*)d_ws;
    const size_t DINV_BYTES = ((size_t)NTOKEN * 4 + 255) & ~(size_t)255;
    const size_t H_BYTES    = (size_t)NTOKEN * NHID * 4;
    float* dinv   = (float*)ws;
    float* Hbuf   = (float*)(ws + DINV_BYTES);
    float* Outbuf = (float*)(ws + DINV_BYTES + H_BYTES);

    const int TPB = 256;

    // 1) degree + dinv (edges shared by both layers)
    deg_init_kernel<<<(NTOKEN + TPB - 1) / TPB, TPB, 0, stream>>>(dinv, NTOKEN);
    deg_accum_kernel<<<(N_EDGES + TPB - 1) / TPB, TPB, 0, stream>>>(edst, dinv, N_EDGES);
    dinv_kernel<<<(NTOKEN + TPB - 1) / TPB, TPB, 0, stream>>>(dinv, NTOKEN);

    // 2) layer 1 GEMM: h0 = emb @ W1   [100000 x 128], K=64
    {
        dim3 blk(32, NHID / 16);                  // 8 waves: one per 16-col tile
        dim3 grd(NTOKEN / 16);                    // 6250 row tiles
        gemm_wmma_f32<NINP, NHID><<<grd, blk, 0, stream>>>(emb, W1, Hbuf);
    }
    // 3) out1 = dinv^2 * h0 + b1, then edge scatter
    {
        long total = (long)NTOKEN * NHID;
        init_out_kernel<<<(int)((total + TPB - 1) / TPB), TPB, 0, stream>>>(
            Hbuf, dinv, b1, Outbuf, NTOKEN, NHID);
        int waves_per_blk = TPB / 32;
        scatter_add_kernel<<<(N_EDGES + waves_per_blk - 1) / waves_per_blk, TPB, 0, stream>>>(
            Hbuf, dinv, esrc, edst, Outbuf, N_EDGES, NHID);
    }

    // 4) layer 2 GEMM: h1 = out1 @ W2  [100000 x 64], K=128  (h1 reuses Hbuf)
    {
        dim3 blk(32, NINP / 16);                  // 4 waves
        dim3 grd(NTOKEN / 16);
        gemm_wmma_f32<NHID, NINP><<<grd, blk, 0, stream>>>(Outbuf, W2, Hbuf);
    }
    // 5) out2 = dinv^2 * h1 + b2, then edge scatter (out2 reuses Outbuf)
    {
        long total = (long)NTOKEN * NINP;
        init_out_kernel<<<(int)((total + TPB - 1) / TPB), TPB, 0, stream>>>(
            Hbuf, dinv, b2, Outbuf, NTOKEN, NINP);
        int waves_per_blk = TPB / 32;
        scatter_add_kernel<<<(N_EDGES + waves_per_blk - 1) / waves_per_blk, TPB, 0, stream>>>(
            Hbuf, dinv, esrc, edst, Outbuf, N_EDGES, NINP);
    }

    // 6) gather: out[b,t,:] = out2[input[b,t],:]
    gather_rows_kernel<<<BATCH * MAX_LEN, NINP, 0, stream>>>(
        Outbuf, inp, out, BATCH * MAX_LEN, NINP);
}